// MinSimilarityScorer_80049600463387
// MI455X (gfx1250) — compile-verified
//
#include <hip/hip_runtime.h>
#include <hip/hip_bf16.h>
#include <float.h>

// Problem constants (match reference)
constexpr int B  = 32;
constexpr int K  = 10;
constexpr int T  = 128;
constexpr int L  = 128;
constexpr int D  = 1024;
constexpr int NT = 32;
constexpr int S  = K * L;      // 1280 support vectors per batch

// NN-kernel tiling
constexpr int NR      = 8;         // S split into 8 ranges across blocks
constexpr int SR      = S / NR;    // 160 support rows per range
constexpr int TILES   = SR / 16;   // 10 16-column tiles per range
constexpr int CD      = 256;       // chunk depth (floats of D staged per step)
constexpr int CH      = D / CD;    // 4 chunks per tile
constexpr int LSTRIDE = 260;       // padded LDS row stride (dwords): 1040B = 65*16
                                   //  -> b128-aligned rows, conflict-free b32 reads

typedef __attribute__((ext_vector_type(2))) float v2f;
typedef __attribute__((ext_vector_type(8))) float v8f;

// ---------------------------------------------------------------------------
// CDNA5 async global->LDS copy (ASYNCcnt-tracked), per CDNA5 ISA
// ---------------------------------------------------------------------------
__device__ __forceinline__ void async_copy_b128(uint32_t lds_byte_off, uint64_t gaddr) {
    asm volatile("global_load_async_to_lds_b128 %0, %1, off"
                 :: "v"(lds_byte_off), "v"(gaddr)
                 : "memory");
}
__device__ __forceinline__ void wait_async_le4() {
    asm volatile("s_wait_asynccnt 0x4" ::: "memory");
}
__device__ __forceinline__ void wait_async_zero() {
    asm volatile("s_wait_asynccnt 0x0" ::: "memory");
}

// ---------------------------------------------------------------------------
// Kernel 1: test_mean[b,t,d] = mean_k test_reps[b,k,t,d]   (float4 per thread)
// ---------------------------------------------------------------------------
__global__ __launch_bounds__(256) void mean_kernel(const float* __restrict__ x,
                                                   float* __restrict__ y) {
    size_t i = ((size_t)blockIdx.x * blockDim.x + threadIdx.x) * 4;
    if (i >= (size_t)B * T * D) return;
    size_t d  = i % D;
    size_t bt = i / D;
    size_t t  = bt % T;
    size_t b  = bt / T;
    float4 s = make_float4(0.f, 0.f, 0.f, 0.f);
    #pragma unroll
    for (int k = 0; k < K; ++k) {
        const float4 q = *(const float4*)(x + (((size_t)b * K + k) * T + t) * D + d);
        s.x += q.x; s.y += q.y; s.z += q.z; s.w += q.w;
    }
    const float inv = 1.0f / (float)K;
    float4 o = make_float4(s.x * inv, s.y * inv, s.z * inv, s.w * inv);
    *(float4*)(y + i) = o;
}

// ---------------------------------------------------------------------------
// Kernel 2: snorm[b,s] = |support_reps[b,s,:]|^2  (one wave32 per row)
// ---------------------------------------------------------------------------
__global__ __launch_bounds__(256) void snorm_kernel(const float* __restrict__ sup,
                                                    float* __restrict__ snorm) {
    const int wave = threadIdx.x >> 5;
    const int lane = threadIdx.x & 31;
    size_t row = (size_t)blockIdx.x * 8 + wave;
    if (row >= (size_t)B * S) return;
    const float* r = sup + row * (size_t)D;
    float acc = 0.f;
    for (int d = lane * 4; d < D; d += 32 * 4) {
        float4 q = *(const float4*)(r + d);
        acc += q.x * q.x + q.y * q.y + q.z * q.z + q.w * q.w;
    }
    #pragma unroll
    for (int off = 16; off >= 1; off >>= 1) acc += __shfl_xor(acc, off, 32);
    if (lane == 0) snorm[row] = acc;
}

// ---------------------------------------------------------------------------
// Kernel 3: prototypes. block = (tag n, batch b); thread owns 4 consecutive d.
// ---------------------------------------------------------------------------
__global__ __launch_bounds__(256) void proto_kernel(const float* __restrict__ sup,
                                                    const float* __restrict__ tgt,
                                                    float* __restrict__ proto) {
    __shared__ float w[S];
    const int n = blockIdx.x, b = blockIdx.y, tid = threadIdx.x;
    for (int s = tid; s < S; s += 256)
        w[s] = tgt[((size_t)b * S + s) * NT + n];
    __syncthreads();
    float4 acc = make_float4(0.f, 0.f, 0.f, 0.f);
    float cnt = 0.f;
    const int d = tid * 4;
    for (int s = 0; s < S; ++s) {
        const float ws = w[s];
        cnt += ws;
        if (ws != 0.0f) {
            float4 q = *(const float4*)(sup + ((size_t)b * S + s) * D + d);
            acc.x += ws * q.x; acc.y += ws * q.y;
            acc.z += ws * q.z; acc.w += ws * q.w;
        }
    }
    const float inv = 1.0f / (cnt + 0.0001f);
    float4 o = make_float4(acc.x * inv, acc.y * inv, acc.z * inv, acc.w * inv);
    *(float4*)(proto + ((size_t)b * NT + n) * D + d) = o;
}

// ---------------------------------------------------------------------------
// Kernel 4: WMMA distance GEMM + partial argmin with async-LDS staging.
// grid = (NR, B), block = 256 (8 waves; wave w owns t-tile w).
// Support chunks (16 rows x 256 floats) are staged into a double-buffered LDS
// tile with global_load_async_to_lds_b128 and consumed by all 8 waves
// (8x less global support traffic). Pipeline: issue chunk g+1, wait ASYNCcnt
// for chunk g, barrier, compute 64 WMMA K-steps, barrier.
// ---------------------------------------------------------------------------
__global__ __launch_bounds__(256) void nn_argmin_kernel(const float* __restrict__ test_mean,
                                                        const float* __restrict__ sup,
                                                        const float* __restrict__ snorm,
                                                        float* __restrict__ pval,
                                                        int* __restrict__ pidx) {
    __shared__ __align__(16) float buf[2][16 * LSTRIDE];   // 33,280 B

    const int r    = blockIdx.x;           // s-range
    const int b    = blockIdx.y;
    const int tid  = threadIdx.x;
    const int wave = tid >> 5;
    const int lane = tid & 31;
    const int n    = lane & 15;            // A row / B column / C column
    const int half = lane >> 4;            // K striping group
    const int t0   = wave * 16;
    const int sbase = r * SR;

    const float* Abase = test_mean + ((size_t)b * T + t0 + n) * D;
    const uint32_t lds0 = (uint32_t)(uintptr_t)&buf[0][0];

    float minv[8];
    int   mini[8];
    #pragma unroll
    for (int v = 0; v < 8; ++v) { minv[v] = FLT_MAX; mini[v] = 0; }

    // cooperative async stage of chunk g into LDS buffer p
    auto stage = [&](int g, int p) {
        const int st = g >> 2, c = g & 3;
        const int s0 = sbase + st * 16;
        const int dbase = c * CD;
        #pragma unroll
        for (int j = 0; j < 4; ++j) {                 // 1024 b128 / 256 threads
            const int e    = tid + 256 * j;
            const int row  = e >> 6;                  // 64 x b128 per row
            const int colq = e & 63;
            const uint64_t ga =
                (uint64_t)(uintptr_t)(sup + ((size_t)b * S + s0 + row) * D + dbase + colq * 4);
            const uint32_t la =
                lds0 + (uint32_t)(p * (16 * LSTRIDE * 4) + row * (LSTRIDE * 4) + colq * 16);
            async_copy_b128(la, ga);
        }
    };

    v8f acc = {};
    stage(0, 0);

    const int G = TILES * CH;                          // 40 chunks
    for (int g = 0; g < G; ++g) {
        const int p  = g & 1;
        const int c  = g & 3;
        const int st = g >> 2;

        if (g + 1 < G) { stage(g + 1, p ^ 1); wait_async_le4(); }
        else           { wait_async_zero(); }
        __syncthreads();                               // chunk g visible to all waves

        if (c == 0) { v8f z = {}; acc = z; }

        const float* Ab   = Abase + c * CD;
        const float* lrow = &buf[p][(size_t)n * LSTRIDE];
        #pragma unroll 4
        for (int dl = 0; dl < CD; dl += 4) {
            // A 16x4 f32: VGPR0/1 hold K = 2*half, 2*half+1 -> contiguous float2
            v2f a = *(const v2f*)(Ab + dl + 2 * half);
            // B 4x16 f32 from LDS: lane needs K = half and K = half+2 of its column
            v2f bb;
            bb.x = lrow[dl + half];
            bb.y = lrow[dl + 2 + half];
            acc = __builtin_amdgcn_wmma_f32_16x16x4_f32(
                false, a, false, bb, (short)0, acc, false, false);
        }

        if (c == 3) {                                  // tile st complete
            const int s0 = sbase + st * 16;
            const float sn = snorm[(size_t)b * S + s0 + n];
            #pragma unroll
            for (int v = 0; v < 8; ++v) {
                const float cand = sn - 2.0f * acc[v];
                if (cand < minv[v]) { minv[v] = cand; mini[v] = s0 + n; }
            }
        }
        __syncthreads();                               // protect buf[p] before re-stage
    }

    // Reduce over the 16 columns held by each half-wave; write range partials.
    #pragma unroll
    for (int v = 0; v < 8; ++v) {
        float bv = minv[v];
        int   bi = mini[v];
        #pragma unroll
        for (int off = 1; off < 16; off <<= 1) {
            const float ov = __shfl_xor(bv, off, 32);
            const int   oi = __shfl_xor(bi, off, 32);
            if (ov < bv || (ov == bv && oi < bi)) { bv = ov; bi = oi; }
        }
        if (n == 0) {
            const size_t o = ((size_t)b * NR + r) * T + t0 + v + 8 * half;
            pval[o] = bv;
            pidx[o] = bi;
        }
    }
}

// ---------------------------------------------------------------------------
// Kernel 4b: combine the NR range-partials (ascending s -> strict < keeps
// the first minimum, matching the torch tie-break).
// ---------------------------------------------------------------------------
__global__ __launch_bounds__(256) void nn_combine_kernel(const float* __restrict__ pval,
                                                         const int* __restrict__ pidx,
                                                         int* __restrict__ nn_idx) {
    const int i = blockIdx.x * blockDim.x + threadIdx.x;   // b*T + t
    if (i >= B * T) return;
    const int b = i / T, t = i % T;
    float best = FLT_MAX;
    int   bi   = 0;
    #pragma unroll
    for (int r = 0; r < NR; ++r) {
        const size_t o = ((size_t)b * NR + r) * T + t;
        const float v = pval[o];
        const int   x = pidx[o];
        if (v < best) { best = v; bi = x; }
    }
    nn_idx[i] = bi;
}

// ---------------------------------------------------------------------------
// Kernel 5: WMMA prototype similarity + fused gather epilogue.
// ---------------------------------------------------------------------------
__global__ __launch_bounds__(32) void proto_sim_kernel(const float* __restrict__ test_mean,
                                                       const float* __restrict__ proto,
                                                       const float* __restrict__ tgt,
                                                       const int* __restrict__ nn_idx,
                                                       float* __restrict__ out) {
    const int b    = blockIdx.y;
    const int t0   = blockIdx.x * 16;
    const int lane = threadIdx.x;
    const int n    = lane & 15;
    const int half = lane >> 4;

    const float* Abase = test_mean + ((size_t)b * T + t0 + n) * D;
    const float* B0 = proto + ((size_t)b * NT + n) * D;
    const float* B1 = proto + ((size_t)b * NT + 16 + n) * D;

    v8f acc0 = {}, acc1 = {};
    #pragma unroll 4
    for (int d = 0; d < D; d += 4) {
        v2f a = *(const v2f*)(Abase + d + 2 * half);
        float4 q0 = *(const float4*)(B0 + d);
        float4 q1 = *(const float4*)(B1 + d);
        v2f b0; b0.x = half ? q0.y : q0.x; b0.y = half ? q0.w : q0.z;
        v2f b1; b1.x = half ? q1.y : q1.x; b1.y = half ? q1.w : q1.z;
        acc0 = __builtin_amdgcn_wmma_f32_16x16x4_f32(
            false, a, false, b0, (short)0, acc0, false, false);
        acc1 = __builtin_amdgcn_wmma_f32_16x16x4_f32(
            false, a, false, b1, (short)0, acc1, false, false);
    }

    #pragma unroll
    for (int v = 0; v < 8; ++v) {
        const int t  = t0 + v + 8 * half;
        const int nn = nn_idx[(size_t)b * T + t];
        const float* trow = tgt + ((size_t)b * S + nn) * (size_t)NT;
        out[((size_t)b * T + t) * NT + n]      = trow[n]      + 0.5f * acc0[v];
        out[((size_t)b * T + t) * NT + 16 + n] = trow[16 + n] + 0.5f * acc1[v];
    }
}

// ---------------------------------------------------------------------------
// Launch
// ---------------------------------------------------------------------------
extern "C" void kernel_launch(void* const* d_in, const int* in_sizes, int n_in,
                              void* d_out, int out_size, void* d_ws, size_t ws_size,
                              hipStream_t stream) {
    const float* test_reps       = (const float*)d_in[0];  // (B,K,T,D)
    const float* support_reps    = (const float*)d_in[1];  // (B,K,L,D) == (B,S,D)
    const float* support_targets = (const float*)d_in[4];  // (B,S,NT)

    float* out_sim   = (float*)d_out;                      // (B,T,NT)
    float* out_proto = out_sim + (size_t)B * T * NT;       // (B,NT,D)

    // Workspace layout
    float* test_mean = (float*)d_ws;                       // B*T*D floats
    float* snorm     = test_mean + (size_t)B * T * D;      // B*S floats
    float* pval      = snorm + (size_t)B * S;              // B*NR*T floats
    int*   pidx      = (int*)(pval + (size_t)B * NR * T);  // B*NR*T ints
    int*   nn        = pidx + (size_t)B * NR * T;          // B*T ints

    {   // test_mean
        const size_t threads = (size_t)B * T * D / 4;
        mean_kernel<<<(unsigned)((threads + 255) / 256), 256, 0, stream>>>(test_reps, test_mean);
    }
    {   // support row norms
        const size_t rows = (size_t)B * S;
        snorm_kernel<<<(unsigned)((rows + 7) / 8), 256, 0, stream>>>(support_reps, snorm);
    }
    {   // prototypes (second output)
        proto_kernel<<<dim3(NT, B), 256, 0, stream>>>(support_reps, support_targets, out_proto);
    }
    {   // WMMA distance GEMM + partial argmin (async-LDS staged)
        nn_argmin_kernel<<<dim3(NR, B), 256, 0, stream>>>(test_mean, support_reps, snorm,
                                                          pval, pidx);
    }
    {   // combine range partials -> nn index
        nn_combine_kernel<<<(B * T + 255) / 256, 256, 0, stream>>>(pval, pidx, nn);
    }
    {   // WMMA proto similarity + fused gather epilogue (first output)
        proto_sim_kernel<<<dim3(T / 16, B), 32, 0, stream>>>(test_mean, out_proto,
                                                             support_targets, nn, out_sim);
    }
}